// DCAB_3435973837522
// MI455X (gfx1250) — compile-verified
//
#include <hip/hip_runtime.h>
#include <hip/hip_bf16.h>

// ---------------------------------------------------------------------------
// Problem constants (from reference)
// ---------------------------------------------------------------------------
#define BATCH  2
#define CC     256
#define HQ     64
#define WQ     64
#define NQ     (HQ * WQ)          // 4096
#define NHEADS 4
#define NLEV   3
#define NPTS   4
#define TAU0   1.0f

typedef __attribute__((ext_vector_type(2))) float v2f;
typedef __attribute__((ext_vector_type(8))) float v8f;

// ---------------------------------------------------------------------------
// fp32 WMMA GEMM:  O[m,n] = epilogue( sum_k A[m,k]*W[n,k] + bias[n] )
//
//  * one wave computes a 32x64 output slab: 2 M-tiles x 4 N-tiles,
//    8 x V_WMMA_F32_16X16X4_F32 per K-step; each B fragment feeds 2 WMMAs,
//    each A fragment feeds 4 -> ~1.33 wmma per load, waits amortized
//  * ACS template: 0 = A row-major (aligned v2f load), else compile-time
//    column stride so strided A taps are immediate-offset b32 loads
//  * EPI: 0 = bias, 1 = bias+exact GELU, 2 = bias+residual
//  * wave-uniform early exit keeps EXEC all-1s around WMMA
// ---------------------------------------------------------------------------
template <int EPI, int ACS>
__global__ __launch_bounds__(128) void dcab_gemm_wmma_f32(
    const float* __restrict__ A, const float* __restrict__ W,
    const float* __restrict__ bias, const float* __restrict__ R,
    float* __restrict__ O,
    int M, int N, int K,
    long aB, int aRS,
    long rB, int rRS, int rCS,
    long oB, int oRS, int oCS,
    int nBatch)
{
    const int lane = threadIdx.x & 31;
    const int wave = threadIdx.x >> 5;
    const int tilesM = M >> 5;                 // 32-tall M slabs
    const int tilesN = N >> 6;                 // 64-wide N slabs
    const long tilesPerBatch = (long)tilesM * tilesN;
    const long total = tilesPerBatch * nBatch;
    const long tile = (long)blockIdx.x * 4 + wave;
    if (tile >= total) return;                 // wave-uniform

    const int b  = (int)(tile / tilesPerBatch);
    const int t  = (int)(tile % tilesPerBatch);
    const int tn = t % tilesN;
    const int tm = t / tilesN;
    const int m0 = tm << 5;
    const int n0 = tn << 6;

    const float* Ab = A + (long)b * aB;
    const float* Rb = (EPI == 2) ? (R + (long)b * rB) : nullptr;
    float*       Ob = O + (long)b * oB;

    const int half = lane >> 4;                // selects K pair {0,1} vs {2,3}
    const int l15  = lane & 15;

    // bases with the half-wave K offset folded in
    const long cs = (ACS == 0) ? 1 : ACS;
    const float* arow0 = Ab + (long)(m0 + l15) * aRS + (long)(half * 2) * cs;
    const float* arow1 = arow0 + (long)16 * aRS;
    const float* w0 = W + (long)(n0 + l15) * K + half * 2;
    const float* w1 = w0 + (long)16 * K;
    const float* w2 = w0 + (long)32 * K;
    const float* w3 = w0 + (long)48 * K;

    v8f c00 = {}, c01 = {}, c02 = {}, c03 = {};
    v8f c10 = {}, c11 = {}, c12 = {}, c13 = {};
#pragma unroll 2
    for (int k0 = 0; k0 < K; k0 += 4) {
        v2f a0, a1;
        if (ACS == 0) {
            a0 = *(const v2f*)(arow0 + k0);             // 8B-aligned b64
            a1 = *(const v2f*)(arow1 + k0);
        } else {                                        // immediate-offset b32
            a0.x = arow0[(long)k0 * ACS];
            a0.y = arow0[(long)(k0 + 1) * ACS];
            a1.x = arow1[(long)k0 * ACS];
            a1.y = arow1[(long)(k0 + 1) * ACS];
        }
        const v2f bf0 = *(const v2f*)(w0 + k0);         // 8B-aligned b64
        const v2f bf1 = *(const v2f*)(w1 + k0);
        const v2f bf2 = *(const v2f*)(w2 + k0);
        const v2f bf3 = *(const v2f*)(w3 + k0);
        c00 = __builtin_amdgcn_wmma_f32_16x16x4_f32(false, a0, false, bf0,
                                                    (short)0, c00, false, false);
        c10 = __builtin_amdgcn_wmma_f32_16x16x4_f32(false, a1, false, bf0,
                                                    (short)0, c10, false, false);
        c01 = __builtin_amdgcn_wmma_f32_16x16x4_f32(false, a0, false, bf1,
                                                    (short)0, c01, false, false);
        c11 = __builtin_amdgcn_wmma_f32_16x16x4_f32(false, a1, false, bf1,
                                                    (short)0, c11, false, false);
        c02 = __builtin_amdgcn_wmma_f32_16x16x4_f32(false, a0, false, bf2,
                                                    (short)0, c02, false, false);
        c12 = __builtin_amdgcn_wmma_f32_16x16x4_f32(false, a1, false, bf2,
                                                    (short)0, c12, false, false);
        c03 = __builtin_amdgcn_wmma_f32_16x16x4_f32(false, a0, false, bf3,
                                                    (short)0, c03, false, false);
        c13 = __builtin_amdgcn_wmma_f32_16x16x4_f32(false, a1, false, bf3,
                                                    (short)0, c13, false, false);
    }

#pragma unroll
    for (int mt = 0; mt < 2; ++mt) {
#pragma unroll
        for (int j = 0; j < 4; ++j) {
            const v8f c =
                (mt == 0)
                    ? ((j == 0) ? c00 : (j == 1) ? c01 : (j == 2) ? c02 : c03)
                    : ((j == 0) ? c10 : (j == 1) ? c11 : (j == 2) ? c12 : c13);
            const int n = n0 + j * 16 + l15;
            const float bv = bias[n];
#pragma unroll
            for (int r = 0; r < 8; ++r) {
                const int m = m0 + mt * 16 + r + half * 8;
                float v = c[r] + bv;
                if (EPI == 1)
                    v = 0.5f * v * (1.0f + erff(v * 0.70710678118654752f));
                if (EPI == 2) v += Rb[(long)m * rRS + (long)n * rCS];
                Ob[(long)m * oRS + (long)n * oCS] = v;
            }
        }
    }
}

// ---------------------------------------------------------------------------
// In-place row LayerNorm over 256 columns. One block (256 threads) per row.
// ---------------------------------------------------------------------------
__global__ __launch_bounds__(256) void dcab_layernorm256(
    float* __restrict__ X, const float* __restrict__ g,
    const float* __restrict__ bta)
{
    __shared__ float buf[256];
    const long row = blockIdx.x;
    const int tid = threadIdx.x;
    const float x = X[row * 256 + tid];

    buf[tid] = x;
    __syncthreads();
    for (int s = 128; s > 0; s >>= 1) {
        if (tid < s) buf[tid] += buf[tid + s];
        __syncthreads();
    }
    const float mu = buf[0] * (1.0f / 256.0f);
    __syncthreads();

    const float d = x - mu;
    buf[tid] = d * d;
    __syncthreads();
    for (int s = 128; s > 0; s >>= 1) {
        if (tid < s) buf[tid] += buf[tid + s];
        __syncthreads();
    }
    const float var = buf[0] * (1.0f / 256.0f);

    X[row * 256 + tid] = d * rsqrtf(var + 1e-5f) * g[tid] + bta[tid];
}

// ---------------------------------------------------------------------------
// Offset / attention-weight predictor + tau-scaled softmax.
// One block per (b,query); Q row cached in LDS.
//   off : [B,Nq, H*L*M*2] = 96      wts : [B,Nq, H*L*M] = 48 (post-softmax)
// ---------------------------------------------------------------------------
__global__ __launch_bounds__(256) void dcab_offwgt(
    const float* __restrict__ Q,
    const float* __restrict__ off_w, const float* __restrict__ off_b,
    const float* __restrict__ wgt_w, const float* __restrict__ wgt_b,
    const float* __restrict__ sim,
    float* __restrict__ OFF, float* __restrict__ WTS)
{
    const long n = blockIdx.x;                 // b*NQ + q
    const int tid = threadIdx.x;
    __shared__ float q[256];
    __shared__ float sw[48];

    q[tid] = Q[n * 256 + tid];
    __syncthreads();

    if (tid < 96) {
        const float* w = off_w + tid * 256;
        float acc = off_b[tid];
        for (int k = 0; k < 256; ++k) acc += q[k] * w[k];
        OFF[n * 96 + tid] = acc;
    } else if (tid < 144) {
        const int j = tid - 96;
        const float* w = wgt_w + j * 256;
        float acc = wgt_b[j];
        for (int k = 0; k < 256; ++k) acc += q[k] * w[k];
        // wts / tau, tau = TAU0/(sim+0.001)  ->  wts * (sim+0.001)/TAU0
        sw[j] = acc * (sim[n] + 0.001f) * (1.0f / TAU0);
    }
    __syncthreads();

    if (tid < NHEADS) {
        float mx = -1e30f;
        for (int i = 0; i < NLEV * NPTS; ++i) mx = fmaxf(mx, sw[tid * 12 + i]);
        float e[NLEV * NPTS];
        float sum = 0.0f;
        for (int i = 0; i < NLEV * NPTS; ++i) {
            e[i] = expf(sw[tid * 12 + i] - mx);
            sum += e[i];
        }
        const float inv = 1.0f / sum;
        for (int i = 0; i < NLEV * NPTS; ++i)
            WTS[n * 48 + tid * 12 + i] = e[i] * inv;
    }
}

// ---------------------------------------------------------------------------
// Deformable bilinear sampling + weighted accumulation.
// Block = (query, head, batch); 256 threads = channels.
// V_l stored channel-last [B, HW_l, 256] -> every tap is a coalesced 1KB read.
// 12 threads precompute tap indices/weights (tanh warp, bilinear weights,
// zero-padding validity, softmax weight folded in) into LDS.
// ---------------------------------------------------------------------------
__global__ __launch_bounds__(256) void dcab_sample(
    const float* __restrict__ V0, const float* __restrict__ V1,
    const float* __restrict__ V2,
    const float* __restrict__ OFF, const float* __restrict__ WTS,
    float* __restrict__ ACC)
{
    const int qn = blockIdx.x;                 // 0..4095
    const int h  = blockIdx.y;                 // 0..3
    const int b  = blockIdx.z;                 // 0..1
    const int tid = threadIdx.x;               // channel
    const long n = (long)b * NQ + qn;

    __shared__ int   sidx[NLEV * NPTS][4];
    __shared__ float swt[NLEV * NPTS][4];

    if (tid < NLEV * NPTS) {
        const int l = tid >> 2;
        const int m = tid & 3;
        const int Wl = (l == 0) ? 64 : ((l == 1) ? 32 : 16);
        const int Hl = Wl;

        const float w  = WTS[n * 48 + h * 12 + tid];
        const float ox = OFF[n * 96 + (((h * NLEV + l) * NPTS + m) * 2) + 0];
        const float oy = OFF[n * 96 + (((h * NLEV + l) * NPTS + m) * 2) + 1];

        const int ix = qn & (WQ - 1);
        const int iy = qn >> 6;
        const float px = (ix + 0.5f) * (1.0f / WQ);
        const float py = (iy + 0.5f) * (1.0f / HQ);

        float gx = tanhf(2.0f * (px + ox) - 1.0f);
        float gy = tanhf(2.0f * (py + oy) - 1.0f);
        gx = fminf(fmaxf(gx, -1.0f), 1.0f);
        gy = fminf(fmaxf(gy, -1.0f), 1.0f);

        const float x = (gx + 1.0f) * 0.5f * (float)(Wl - 1);
        const float y = (gy + 1.0f) * 0.5f * (float)(Hl - 1);
        const float x0f = floorf(x), y0f = floorf(y);
        const int x0 = (int)x0f, y0 = (int)y0f;
        const float wx1 = x - x0f, wy1 = y - y0f;
        const float wx0 = 1.0f - wx1, wy0 = 1.0f - wy1;

        const int base = b * Hl * Wl;
        const int   xs[2]  = {x0, x0 + 1};
        const int   ys_[2] = {y0, y0 + 1};
        const float wxs[2] = {wx0, wx1};
        const float wys[2] = {wy0, wy1};
#pragma unroll
        for (int t = 0; t < 4; ++t) {
            const int yi = ys_[t >> 1], xi = xs[t & 1];
            const bool valid = (xi >= 0) && (xi < Wl) && (yi >= 0) && (yi < Hl);
            const int yc = min(max(yi, 0), Hl - 1);
            const int xc = min(max(xi, 0), Wl - 1);
            sidx[tid][t] = (base + yc * Wl + xc) * CC;
            swt[tid][t]  = valid ? (w * wys[t >> 1] * wxs[t & 1]) : 0.0f;
        }
    }
    __syncthreads();

    float acc = 0.0f;
#pragma unroll
    for (int e = 0; e < NLEV * NPTS; ++e) {
        const int l = e >> 2;
        const float* vp = (l == 0) ? V0 : ((l == 1) ? V1 : V2);
#pragma unroll
        for (int t = 0; t < 4; ++t)
            acc += swt[e][t] * vp[sidx[e][t] + tid];
    }
    // ACC layout [B, Nq, H*C]
    ACC[(n * NHEADS + h) * CC + tid] = acc;
}

// ---------------------------------------------------------------------------
// Launch
// ---------------------------------------------------------------------------
static inline int gemm_blocks(int M, int N, int nb)
{
    long tiles = (long)(M >> 5) * (N >> 6) * nb;   // 32x64 slabs, 4 waves/blk
    return (int)((tiles + 3) / 4);
}

extern "C" void kernel_launch(void* const* d_in, const int* in_sizes, int n_in,
                              void* d_out, int out_size, void* d_ws, size_t ws_size,
                              hipStream_t stream)
{
    const float* S     = (const float*)d_in[0];
    const float* f0    = (const float*)d_in[1];
    const float* f1    = (const float*)d_in[2];
    const float* f2    = (const float*)d_in[3];
    const float* sim   = (const float*)d_in[4];
    const float* vw0   = (const float*)d_in[5];
    const float* vb0   = (const float*)d_in[6];
    const float* vw1   = (const float*)d_in[7];
    const float* vb1   = (const float*)d_in[8];
    const float* vw2   = (const float*)d_in[9];
    const float* vb2   = (const float*)d_in[10];
    const float* q_w   = (const float*)d_in[11];
    const float* q_b   = (const float*)d_in[12];
    const float* off_w = (const float*)d_in[13];
    const float* off_b = (const float*)d_in[14];
    const float* wgt_w = (const float*)d_in[15];
    const float* wgt_b = (const float*)d_in[16];
    const float* out_w = (const float*)d_in[17];
    const float* out_b = (const float*)d_in[18];
    const float* lnq_g = (const float*)d_in[19];
    const float* lnq_b = (const float*)d_in[20];
    const float* lno_g = (const float*)d_in[21];
    const float* lno_b = (const float*)d_in[22];
    const float* fc1_w = (const float*)d_in[23];
    const float* fc1_b = (const float*)d_in[24];
    const float* fc2_w = (const float*)d_in[25];
    const float* fc2_b = (const float*)d_in[26];
    float* out = (float*)d_out;

    // workspace carve-up (floats); ~100 MB total
    float* w = (float*)d_ws;
    float* V0 = w;  w += (long)BATCH * NQ * CC;          // [B,4096,256]
    float* V1 = w;  w += (long)BATCH * 1024 * CC;        // [B,1024,256]
    float* V2 = w;  w += (long)BATCH * 256 * CC;         // [B, 256,256]
    float* Q  = w;  w += (long)BATCH * NQ * CC;          // [B,4096,256]
    float* OF = w;  w += (long)BATCH * NQ * 96;          // offsets
    float* WT = w;  w += (long)BATCH * NQ * 48;          // softmax weights
    float* AC = w;  w += (long)BATCH * NQ * NHEADS * CC; // [B,4096,1024]
    float* Z  = w;  w += (long)BATCH * NQ * CC;          // [B,4096,256]
    float* HB = w;  w += (long)BATCH * NQ * 4 * CC;      // [B,4096,1024]

    // 1) Value projections (NCHW feature maps read transposed, write [HW,C])
    dcab_gemm_wmma_f32<0, NQ><<<gemm_blocks(NQ, CC, BATCH), 128, 0, stream>>>(
        f0, vw0, vb0, nullptr, V0, NQ, CC, 256,
        (long)256 * NQ, 1,   0, 0, 0,   (long)NQ * CC, CC, 1, BATCH);
    dcab_gemm_wmma_f32<0, 1024><<<gemm_blocks(1024, CC, BATCH), 128, 0, stream>>>(
        f1, vw1, vb1, nullptr, V1, 1024, CC, 512,
        (long)512 * 1024, 1, 0, 0, 0, (long)1024 * CC, CC, 1, BATCH);
    dcab_gemm_wmma_f32<0, 256><<<gemm_blocks(256, CC, BATCH), 128, 0, stream>>>(
        f2, vw2, vb2, nullptr, V2, 256, CC, 1024,
        (long)1024 * 256, 1,  0, 0, 0, (long)256 * CC, CC, 1, BATCH);

    // 2) Q = LN(S_seq @ q_w^T + q_b)
    dcab_gemm_wmma_f32<0, NQ><<<gemm_blocks(NQ, CC, BATCH), 128, 0, stream>>>(
        S, q_w, q_b, nullptr, Q, NQ, CC, CC,
        (long)CC * NQ, 1,    0, 0, 0,   (long)NQ * CC, CC, 1, BATCH);
    dcab_layernorm256<<<BATCH * NQ, 256, 0, stream>>>(Q, lnq_g, lnq_b);

    // 3) Offsets + tau-scaled softmax weights
    dcab_offwgt<<<BATCH * NQ, 256, 0, stream>>>(
        Q, off_w, off_b, wgt_w, wgt_b, sim, OF, WT);

    // 4) Deformable sampling -> ACC [B,Nq,H*C]
    dcab_sample<<<dim3(NQ, NHEADS, BATCH), 256, 0, stream>>>(
        V0, V1, V2, OF, WT, AC);

    // 5) Z = LN(S_seq + ACC @ out_w^T + out_b)
    dcab_gemm_wmma_f32<2, 0><<<gemm_blocks(NQ, CC, BATCH), 128, 0, stream>>>(
        AC, out_w, out_b, S, Z, NQ, CC, NHEADS * CC,
        (long)NQ * NHEADS * CC, NHEADS * CC,
        (long)CC * NQ, 1, NQ,
        (long)NQ * CC, CC, 1, BATCH);
    dcab_layernorm256<<<BATCH * NQ, 256, 0, stream>>>(Z, lno_g, lno_b);

    // 6) HB = gelu_exact(Z @ fc1^T + fc1_b)
    dcab_gemm_wmma_f32<1, 0><<<gemm_blocks(NQ, 4 * CC, BATCH), 128, 0, stream>>>(
        Z, fc1_w, fc1_b, nullptr, HB, NQ, 4 * CC, CC,
        (long)NQ * CC, CC,   0, 0, 0,   (long)NQ * 4 * CC, 4 * CC, 1, BATCH);

    // 7) out = transpose(Z + HB @ fc2^T + fc2_b) -> [B,C,Hq,Wq]
    dcab_gemm_wmma_f32<2, 0><<<gemm_blocks(NQ, CC, BATCH), 128, 0, stream>>>(
        HB, fc2_w, fc2_b, Z, out, NQ, CC, 4 * CC,
        (long)NQ * 4 * CC, 4 * CC,
        (long)NQ * CC, CC, 1,
        (long)CC * NQ, 1, NQ, BATCH);
}